// CombinedLoss_42614665511278
// MI455X (gfx1250) — compile-verified
//
#include <hip/hip_runtime.h>
#include <math.h>

typedef __attribute__((ext_vector_type(2))) float v2f;
typedef __attribute__((ext_vector_type(8))) float v8f;

// ---------------------------------------------------------------------------
// Workspace layout:
//   float acc[8]           (acc[0]=S, acc[1..2]=cd means, acc[3..4]=seed means,
//                           acc[5]=confidence mean)
//   unsigned mins[36864]   (bit-patterns of min d2; float bits are monotone
//                           for non-negative values, so atomicMin(u32) works)
//     mA1 = mins+0      [8192] each pc2-flat pt  -> min d2 to pc1_0-flat
//     mA2 = mins+8192   [8192] each pc1_0-flat   -> min d2 to pc2-flat
//     mB1 = mins+16384  [8192] each pc2-flat pt  -> min d2 to pc1_1-flat
//     mB2 = mins+24576  [4096] each pc1_1-flat   -> min d2 to pc2-flat
//     mC  = mins+28672  [8192] each (b,p) pc3 pt -> min d2 to pc2[b,:]
// ---------------------------------------------------------------------------

#define NMINS 36864
#define WSN   (8 + NMINS)

__global__ void init_ws(unsigned* w) {
    int i = blockIdx.x * 256 + threadIdx.x;
    if (i < WSN) w[i] = (i < 8) ? 0u : 0x7F7FFFFFu;  // 0.0f / FLT_MAX bits
}

// Partial NN-min: block (bx) owns 256 query points, block (by) owns a 512-pt
// reference chunk. Per-thread min over the chunk, then one u32 atomicMin.
// All dims are exact multiples -> no bounds checks, no divergence.
__global__ __launch_bounds__(256)
void nn_min_partial(const float* __restrict__ Q,
                    const float* __restrict__ R,  // base of ref cloud
                    int chunk,                    // refs per y-block
                    unsigned* __restrict__ mins) {
    __shared__ float tile[256 * 3];
    int gid = blockIdx.x * 256 + threadIdx.x;
    const float* Rc = R + (size_t)blockIdx.y * chunk * 3;

    float qx = Q[gid * 3 + 0];
    float qy = Q[gid * 3 + 1];
    float qz = Q[gid * 3 + 2];

    float best = 3.4e38f;
    for (int base = 0; base < chunk; base += 256) {
        int j = base + threadIdx.x;
        tile[threadIdx.x * 3 + 0] = Rc[j * 3 + 0];
        tile[threadIdx.x * 3 + 1] = Rc[j * 3 + 1];
        tile[threadIdx.x * 3 + 2] = Rc[j * 3 + 2];
        __syncthreads();
#pragma unroll 4
        for (int k = 0; k < 256; ++k) {
            float dx = qx - tile[k * 3 + 0];
            float dy = qy - tile[k * 3 + 1];
            float dz = qz - tile[k * 3 + 2];
            float d2 = dx * dx + dy * dy + dz * dz;
            best = fminf(best, d2);
        }
        __syncthreads();
    }
    atomicMin(mins + gid, __float_as_uint(best));
}

// Confidence partial min: per-batch reference (pc2[b]). P=2048, 256 | P, so
// each x-block has a uniform batch index.
__global__ __launch_bounds__(256)
void conf_min_partial(const float* __restrict__ pc3,
                      const float* __restrict__ pc2,
                      unsigned* __restrict__ mins) {
    const int P = 2048, M = 2048, CH = 512;
    __shared__ float tile[256 * 3];
    int gid = blockIdx.x * 256 + threadIdx.x;
    int b = gid / P;
    const float* Rc = pc2 + ((size_t)b * M + blockIdx.y * CH) * 3;

    float qx = pc3[gid * 3 + 0];
    float qy = pc3[gid * 3 + 1];
    float qz = pc3[gid * 3 + 2];

    float best = 3.4e38f;
    for (int base = 0; base < CH; base += 256) {
        int j = base + threadIdx.x;
        tile[threadIdx.x * 3 + 0] = Rc[j * 3 + 0];
        tile[threadIdx.x * 3 + 1] = Rc[j * 3 + 1];
        tile[threadIdx.x * 3 + 2] = Rc[j * 3 + 2];
        __syncthreads();
#pragma unroll 4
        for (int k = 0; k < 256; ++k) {
            float dx = qx - tile[k * 3 + 0];
            float dy = qy - tile[k * 3 + 1];
            float dz = qz - tile[k * 3 + 2];
            float d2 = dx * dx + dy * dy + dz * dz;
            best = fminf(best, d2);
        }
        __syncthreads();
    }
    atomicMin(mins + gid, __float_as_uint(best));
}

// Sum sqrt(min d2) * invn into *acc (mean of NN distances).
__global__ __launch_bounds__(256)
void sum_sqrt_min(const unsigned* __restrict__ mins, float invn,
                  float* __restrict__ acc) {
    __shared__ float red[256];
    int gid = blockIdx.x * 256 + threadIdx.x;
    red[threadIdx.x] = __builtin_amdgcn_sqrtf(__uint_as_float(mins[gid]));
    __syncthreads();
    for (int s = 128; s > 0; s >>= 1) {
        if (threadIdx.x < s) red[threadIdx.x] += red[threadIdx.x + s];
        __syncthreads();
    }
    if (threadIdx.x == 0) atomicAdd(acc, red[0] * invn);
}

// Confidence finish: mean of (pc1_3 - exp(-sqrt(min d2)))^2 over 8192 pts.
__global__ __launch_bounds__(256)
void conf_finish(const unsigned* __restrict__ mins,
                 const float* __restrict__ pc13, float* __restrict__ acc) {
    __shared__ float red[256];
    int gid = blockIdx.x * 256 + threadIdx.x;
    float d = __builtin_amdgcn_sqrtf(__uint_as_float(mins[gid]));
    float diff = pc13[gid] - expf(-d);
    red[threadIdx.x] = diff * diff;
    __syncthreads();
    for (int s = 128; s > 0; s >>= 1) {
        if (threadIdx.x < s) red[threadIdx.x] += red[threadIdx.x + s];
        __syncthreads();
    }
    if (threadIdx.x == 0) atomicAdd(acc, red[0] * (1.0f / 8192.0f));
}

__global__ void finalize(float* acc) {
    // acc[1..4] already mean-scaled; ALPHA = 0.5
    acc[0] = acc[5] + 0.5f * (acc[1] + acc[2]) + (acc[3] + acc[4]);
}

// Stage 2: out[b,n,m] = S + 0.5 * sqrt(max(0, |x|^2 + |y|^2 - 2 x.y))
// One 16x16 tile per wave via V_WMMA_F32_16X16X4_F32 (K=3 padded to 4).
__global__ __launch_bounds__(256)
void pairwise_wmma(const float* __restrict__ X, const float* __restrict__ Y,
                   const float* __restrict__ Sptr, float* __restrict__ out) {
    const int N = 2048, M = 2048;
    const int TN = M / 16;           // 128 tiles along M
    const int TM = N / 16;           // 128 tiles along N

    int wave = blockIdx.x * (blockDim.x >> 5) + (threadIdx.x >> 5);
    int lane = threadIdx.x & 31;
    int b    = wave / (TM * TN);
    int rem  = wave % (TM * TN);
    int tm   = rem / TN;
    int tn   = rem % TN;

    int col  = lane & 15;            // M-index for A load, N-index for B/C/D
    int half = lane >> 4;            // 0: K=0,1  1: K=2,3 (A/B layout halves)

    const float* xp = X + (size_t)(b * N + tm * 16 + col) * 3;
    float x0 = xp[0], x1 = xp[1], x2 = xp[2];
    const float* yp = Y + (size_t)(b * M + tn * 16 + col) * 3;
    float y0 = yp[0], y1 = yp[1], y2 = yp[2];

    float xs = x0 * x0 + x1 * x1 + x2 * x2;   // |x_row|^2 (row = col here)
    float ys = y0 * y0 + y1 * y1 + y2 * y2;   // |y_col|^2

    // A (16x4 f32): lanes 0-15 -> {K0,K1}, lanes 16-31 -> {K2,K3=0}
    v2f Am; Am.x = half ? x2 : x0; Am.y = half ? 0.0f : x1;
    // B (4x16 f32): same half-split over K, N striped across lanes
    v2f Bm; Bm.x = half ? y2 : y0; Bm.y = half ? 0.0f : y1;

    v8f acc = {0.f, 0.f, 0.f, 0.f, 0.f, 0.f, 0.f, 0.f};
    acc = __builtin_amdgcn_wmma_f32_16x16x4_f32(
        /*neg_a=*/false, Am, /*neg_b=*/false, Bm,
        /*c_mod=*/(short)0, acc, /*reuse_a=*/false, /*reuse_b=*/false);

    float S = Sptr[0];
#pragma unroll
    for (int v = 0; v < 8; ++v) {
        int row = v + 8 * half;                   // D layout: M = v + 8*(lane>=16)
        float xr = __shfl(xs, row, 32);           // broadcast |x_row|^2
        float d2 = xr + ys - 2.0f * acc[v];
        d2 = d2 > 0.0f ? d2 : 0.0f;
        out[(size_t)(b * N + tm * 16 + row) * M + tn * 16 + col] =
            S + 0.5f * __builtin_amdgcn_sqrtf(d2);
    }
}

extern "C" void kernel_launch(void* const* d_in, const int* in_sizes, int n_in,
                              void* d_out, int out_size, void* d_ws, size_t ws_size,
                              hipStream_t stream) {
    const float* pc1_0 = (const float*)d_in[0];  // [4,2048,3]
    const float* pc1_1 = (const float*)d_in[1];  // [4,1024,3]
    const float* pc1_3 = (const float*)d_in[2];  // [4,2048,1]
    const float* pc2   = (const float*)d_in[3];  // [4,2048,3]
    const float* pc3   = (const float*)d_in[4];  // [4,2048,3]
    float* out = (float*)d_out;                  // [4,2048,2048]

    float*    acc  = (float*)d_ws;
    unsigned* mins = (unsigned*)d_ws + 8;
    unsigned* mA1 = mins;            // 8192
    unsigned* mA2 = mins + 8192;     // 8192
    unsigned* mB1 = mins + 16384;    // 8192
    unsigned* mB2 = mins + 24576;    // 4096
    unsigned* mC  = mins + 28672;    // 8192

    init_ws<<<(WSN + 255) / 256, 256, 0, stream>>>((unsigned*)d_ws);

    // cd = chamfer(pc1_0 flat [8192], pc2 flat [8192]) -- ref dim split 512/chunk
    nn_min_partial<<<dim3(32, 16), 256, 0, stream>>>(pc2,   pc1_0, 512, mA1);
    nn_min_partial<<<dim3(32, 16), 256, 0, stream>>>(pc1_0, pc2,   512, mA2);
    // seed = chamfer(pc1_1 flat [4096], pc2 flat [8192])
    nn_min_partial<<<dim3(32,  8), 256, 0, stream>>>(pc2,   pc1_1, 512, mB1);
    nn_min_partial<<<dim3(16, 16), 256, 0, stream>>>(pc1_1, pc2,   512, mB2);
    // confidence per-batch mins
    conf_min_partial<<<dim3(32, 4), 256, 0, stream>>>(pc3, pc2, mC);

    // Reduce mins to mean NN distances / confidence MSE
    sum_sqrt_min<<<32, 256, 0, stream>>>(mA1, 1.0f / 8192.0f, acc + 1);
    sum_sqrt_min<<<32, 256, 0, stream>>>(mA2, 1.0f / 8192.0f, acc + 2);
    sum_sqrt_min<<<32, 256, 0, stream>>>(mB1, 1.0f / 8192.0f, acc + 3);
    sum_sqrt_min<<<16, 256, 0, stream>>>(mB2, 1.0f / 4096.0f, acc + 4);
    conf_finish <<<32, 256, 0, stream>>>(mC, pc1_3, acc + 5);

    finalize<<<1, 1, 0, stream>>>(acc);

    // 4 * 128 * 128 = 65536 tiles, 8 waves per 256-thread block -> 8192 blocks
    pairwise_wmma<<<8192, 256, 0, stream>>>(pc1_0, pc2, acc, out);
}